// ShoeboxToRIR_69303592288509
// MI455X (gfx1250) — compile-verified
//
#include <hip/hip_runtime.h>
#include <cstdint>

#define MAX_ORDER 15
#define GRID_W    31                 // 2*MAX_ORDER+1
#define NCELLS    (31 * 31 * 31)     // 29791 lattice cells, 4991 valid
#define RIR_LEN   3968
#define WIN_LEN   81
#define PAD       40                 // WIN_LEN/2
#define OUT_LEN   (RIR_LEN - PAD)    // 3928
#define SPLIT     8                  // blocks per batch
#define NTHREADS  256                // 8 wave32s

__device__ __forceinline__ int iabs(int v) { return v < 0 ? -v : v; }

__global__ void zero_out_kernel(float* __restrict__ out, int n) {
    int i = blockIdx.x * blockDim.x + threadIdx.x;
    if (i < n) out[i] = 0.0f;
}

__global__ __launch_bounds__(NTHREADS)
void rir_kernel(const float* __restrict__ latent, float* __restrict__ out, int B) {
    const int b     = blockIdx.x / SPLIT;
    const int split = blockIdx.x % SPLIT;
    const int tid   = threadIdx.x;

    __shared__ float          s_rir[RIR_LEN];   // 15872 B
    __shared__ unsigned short s_list[1024];     // owned lattice cells (~624)
    __shared__ float          s_lat[12];
    __shared__ float          s_pow[6][9];      // tr[j]^e, e=0..8
    __shared__ int            s_count;

    if (tid == 0) s_count = 0;

    // ---- CDNA5 async DMA: global -> LDS load of this batch's latent row ----
    if (tid < 12) {
        const float* gp  = latent + b * 12 + tid;
        unsigned     lds = (unsigned)(uintptr_t)(&s_lat[tid]); // low 32b of generic LDS addr
        asm volatile("global_load_async_to_lds_b32 %0, %1, off"
                     :: "v"(lds), "v"(gp) : "memory");
    }

    for (int j = tid; j < RIR_LEN; j += NTHREADS) s_rir[j] = 0.0f;

    asm volatile("s_wait_asynccnt 0" ::: "memory");   // ASYNCcnt == 0 -> LDS data visible
    __syncthreads();

    // ---- per-batch scalars (uniform across threads) ----
    const float room0 = s_lat[0] + 1.0f, room1 = s_lat[1] + 1.0f, room2 = s_lat[2] + 1.0f;
    const float mic0  = s_lat[3] * room0, mic1 = s_lat[4] * room1, mic2 = s_lat[5] * room2;
    const float src0  = s_lat[6] * room0, src1 = s_lat[7] * room1, src2 = s_lat[8] * room2;

    if (tid < 6) {
        // absorption[0..3]=lat[9], [4]=lat[10], [5]=lat[11]
        float a  = (tid < 4) ? s_lat[9] : s_lat[6 + tid];
        float ab = a * 0.84f + 0.01f;
        float tr = sqrtf(1.0f - ab);
        float p  = 1.0f;
        for (int e = 0; e < 9; ++e) { s_pow[tid][e] = p; p *= tr; }
    }

    // ---- deterministic compaction: this block owns cells c with c%SPLIT==split ----
    for (int k = tid;; k += NTHREADS) {
        int c = split + SPLIT * k;
        if (c >= NCELLS) break;
        int gz = c / 961;  int r  = c - gz * 961;
        int gy = r / 31;   int gx = r - gy * 31;
        gx -= MAX_ORDER; gy -= MAX_ORDER; gz -= MAX_ORDER;
        if (iabs(gx) + iabs(gy) + iabs(gz) <= MAX_ORDER) {
            int pos = atomicAdd(&s_count, 1);          // ds_add_rtn_u32
            s_list[pos] = (unsigned short)c;
        }
    }
    __syncthreads();
    const int count = s_count;

    const float FS_C   = 16000.0f / 343.0f;
    const float INV4PI = 0.07957747154594767f;
    const float INV_PI = 0.3183098861837907f;
    // Hann rotation constants: theta = 2*pi/81
    const float rotC = cospif(2.0f / (float)WIN_LEN);
    const float rotS = sinpif(2.0f / (float)WIN_LEN);

    // ---- main loop: each thread handles whole image sources (uniform 81-tap loop) ----
    for (int i = tid; i < count; i += NTHREADS) {
        int c  = s_list[i];
        int gz = c / 961;  int r  = c - gz * 961;
        int gy = r / 31;   int gx = r - gy * 31;
        gx -= MAX_ORDER; gy -= MAX_ORDER; gz -= MAX_ORDER;

        // Python-modulo odd test == (g & 1) on two's complement
        float img0 = (gx & 1) ? room0 * (float)(gx + 1) - src0 : room0 * (float)gx + src0;
        float img1 = (gy & 1) ? room1 * (float)(gy + 1) - src1 : room1 * (float)gy + src1;
        float img2 = (gz & 1) ? room2 * (float)(gz + 1) - src2 : room2 * (float)gz + src2;

        // |floor(g/2)| and |floor((g+1)/2)| via arithmetic shift (floor division)
        int elox = iabs(gx >> 1), ehix = iabs((gx + 1) >> 1);
        int eloy = iabs(gy >> 1), ehiy = iabs((gy + 1) >> 1);
        int eloz = iabs(gz >> 1), ehiz = iabs((gz + 1) >> 1);
        float att = s_pow[0][elox] * s_pow[1][ehix] *
                    s_pow[2][eloy] * s_pow[3][ehiy] *
                    s_pow[4][eloz] * s_pow[5][ehiz];

        float v0 = img0 - mic0, v1 = img1 - mic1, v2 = img2 - mic2;
        float dist  = sqrtf(v0 * v0 + v1 * v1 + v2 * v2);
        float amp   = att * INV4PI / dist;
        float delay = dist * FS_C;
        float idl   = floorf(delay);
        float fr    = delay - idl;          // in [0,1)
        int   base  = (int)idl;

        // sinc numerator: sin(pi*(m-fr)) = -(-1)^m * sin(pi*fr); sign alternates with m
        float sp    = sinpif(fr);
        float s_val = -sp;                  // m = -PAD (even) -> -sp; negate each tap

        // Hann phase: phi = theta*(m - fr), advanced by fixed rotation each tap
        float t0 = (2.0f / (float)WIN_LEN) * ((float)(-PAD) - fr);
        float hs, hc;
        sincospif(t0, &hs, &hc);            // sin/cos(theta*(-PAD - fr))

        float x = (float)(-PAD) - fr;       // advances by +1.0f per tap

        #pragma unroll 3
        for (int k = 0; k < WIN_LEN; ++k) {
            float sv;
            if (x == 0.0f) sv = 1.0f;                          // sinc(0) = 1
            else           sv = s_val * INV_PI * __builtin_amdgcn_rcpf(x);
            float win = 0.5f + 0.5f * hc;                      // Hann window
            float tap = amp * sv * win;
            int idx = base + k;                                // == i_delay + k (>= 0)
            if ((unsigned)idx < (unsigned)RIR_LEN)
                atomicAdd(&s_rir[idx], tap);                   // ds_add_f32 (no-return)
            // advance: sign flip, phase rotation, x += 1
            s_val = -s_val;
            float hc2 = hc * rotC - hs * rotS;
            float hs2 = hs * rotC + hc * rotS;
            hc = hc2; hs = hs2;
            x += 1.0f;
        }
    }
    __syncthreads();

    // ---- merge LDS partial into global RIR (skip zeros to cut atomic traffic) ----
    float* rb = out + (size_t)b * OUT_LEN;
    for (int j = tid; j < OUT_LEN; j += NTHREADS) {
        float v = s_rir[j + PAD];
        if (v != 0.0f) atomicAdd(&rb[j], v);           // global_atomic_add_f32
    }

    if (split == 0 && tid == 0) {
        float d0 = mic0 - src0, d1 = mic1 - src1, d2 = mic2 - src2;
        out[(size_t)B * OUT_LEN + b] = sqrtf(d0 * d0 + d1 * d1 + d2 * d2) * FS_C;
    }
}

extern "C" void kernel_launch(void* const* d_in, const int* in_sizes, int n_in,
                              void* d_out, int out_size, void* d_ws, size_t ws_size,
                              hipStream_t stream) {
    (void)n_in; (void)d_ws; (void)ws_size;
    const float* latent = (const float*)d_in[0];
    float* out = (float*)d_out;
    const int B = in_sizes[0] / 12;       // (B,12) latent

    // d_out is poisoned by the harness; accumulation needs zeros.
    zero_out_kernel<<<(out_size + 255) / 256, 256, 0, stream>>>(out, out_size);
    rir_kernel<<<B * SPLIT, NTHREADS, 0, stream>>>(latent, out, B);
}